// StateGNNEncoderConvEdgeAttrCompact_42528766165971
// MI455X (gfx1250) — compile-verified
//
#include <hip/hip_runtime.h>
#include <hip/hip_bf16.h>

typedef __attribute__((ext_vector_type(2))) float v2f;
typedef __attribute__((ext_vector_type(8))) float v8f;

#define DEV_FADD(p, v) __hip_atomic_fetch_add((p), (v), __ATOMIC_RELAXED, __HIP_MEMORY_SCOPE_AGENT)

// ---------------------------------------------------------------- utilities

__global__ void fill0_kernel(float* __restrict__ p, long long n) {
    long long i = (long long)blockIdx.x * blockDim.x + threadIdx.x;
    if (i < n) p[i] = 0.0f;
}

__global__ void degree_kernel(const int* __restrict__ dst, float* __restrict__ deg,
                              long long E) {
    long long e = (long long)blockIdx.x * blockDim.x + threadIdx.x;
    if (e < E) DEV_FADD(&deg[dst[e]], 1.0f);
}

// deg -> dinv = deg>0 ? rsqrt(deg) : 0   (in place)
__global__ void rsqrt_deg_kernel(float* __restrict__ d, long long n) {
    long long i = (long long)blockIdx.x * blockDim.x + threadIdx.x;
    if (i < n) {
        float v = d[i];
        d[i] = (v > 0.0f) ? rsqrtf(v) : 0.0f;
    }
}

// ---------------------------------------------------------------- scatters
// MODE 0: symmetric gcn norm, aux = dinv[node]   out[dst] += in[src]*dinv[s]*dinv[d]
// MODE 1: per-edge weight,    aux = w[edge]      out[dst] += in[src]*w[e]
// MODE 2: plain               out[dst] += in[src]
template <int F, int MODE>
__global__ void scatter_kernel(const int* __restrict__ src, const int* __restrict__ dst,
                               const float* __restrict__ xin, float* __restrict__ xout,
                               const float* __restrict__ aux, long long E) {
    long long idx = (long long)blockIdx.x * blockDim.x + threadIdx.x;
    long long e = idx / F;
    int f = (int)(idx % F);
    if (e >= E) return;
    int s = src[e];
    int d = dst[e];
    float v = xin[(size_t)s * F + f];
    float scale;
    if (MODE == 0)      scale = aux[s] * aux[d];
    else if (MODE == 1) scale = aux[e];
    else                scale = 1.0f;
    DEV_FADD(&xout[(size_t)d * F + f], v * scale);
}

// ------------------------------------------------- WMMA GEMM: C (+)= A @ W
// A: [nrows, 64] row-major fp32, W: [64, 64] row-major fp32 (K x N).
// V_WMMA_F32_16X16X4_F32.  Weights staged transposed in LDS with row stride
// 66 (even -> 8B alignment kept for ds_load_b64; stride-64 would put every
// lane on the same bank pair).  A row is loaded once into 32 VGPRs and
// reused across all 4 column tiles.
__global__ void __launch_bounds__(256)
gemm64_wmma(const float* __restrict__ A, const float* __restrict__ W,
            float* __restrict__ C, int nrows, int accumulate) {
    __shared__ float Wt[64 * 66];  // Wt[n*66 + k] = W[k*64 + n]
    for (int i = threadIdx.x; i < 64 * 64; i += 256) {
        int k = i >> 6, n = i & 63;
        Wt[n * 66 + k] = W[i];
    }
    __syncthreads();

    const int lane = threadIdx.x & 31;
    const int wave = threadIdx.x >> 5;
    const int row0 = (blockIdx.x * 8 + wave) * 16;
    if (row0 >= nrows) return;  // wave-uniform, no barriers below

    const int mloc = lane & 15;
    // 16x4 f32 A layout: lanes 0-15 carry K=k0+0/1, lanes 16-31 carry K=k0+2/3
    const int khalf = (lane < 16) ? 0 : 2;
    const int rbase = (lane < 16) ? 0 : 8;  // C/D: VGPR r -> row r (+8 hi lanes)
    const bool fullTile = (row0 + 16 <= nrows);  // wave-uniform

    int mrow = row0 + mloc;
    int mclamp = (mrow < nrows) ? mrow : (nrows - 1);
    const v2f* arow = (const v2f*)(A + (size_t)mclamp * 64);

    v2f areg[16];
    #pragma unroll
    for (int kk = 0; kk < 16; ++kk) areg[kk] = arow[kk * 2 + (khalf >> 1)];

    for (int ct = 0; ct < 4; ++ct) {
        const int ncol = ct * 16 + mloc;
        const v2f* brow = (const v2f*)(&Wt[ncol * 66 + khalf]);
        float* crow = C + (size_t)(row0 + rbase) * 64 + ncol;

        v8f acc;
        if (accumulate) {
            if (fullTile) {
                #pragma unroll
                for (int r = 0; r < 8; ++r) acc[r] = crow[(size_t)r * 64];
            } else {
                #pragma unroll
                for (int r = 0; r < 8; ++r) {
                    int row = row0 + rbase + r;
                    acc[r] = (row < nrows) ? crow[(size_t)r * 64] : 0.0f;
                }
            }
        } else {
            #pragma unroll
            for (int r = 0; r < 8; ++r) acc[r] = 0.0f;
        }

        #pragma unroll
        for (int kk = 0; kk < 16; ++kk) {
            v2f b = brow[kk * 2];  // {B[4kk+khalf][ncol], B[4kk+khalf+1][ncol]}
            acc = __builtin_amdgcn_wmma_f32_16x16x4_f32(false, areg[kk], false, b,
                                                        (short)0, acc, false, false);
        }

        if (fullTile) {
            #pragma unroll
            for (int r = 0; r < 8; ++r) crow[(size_t)r * 64] = acc[r];
        } else {
            #pragma unroll
            for (int r = 0; r < 8; ++r) {
                int row = row0 + rbase + r;
                if (row < nrows) crow[(size_t)r * 64] = acc[r];
            }
        }
    }
}

// ---------------------------------------------------------------- fused ops

// gx = relu(x@W0 + h1@W1 + h2@W2 + b), x/h: [NV,5], W: [3,5,64]
__global__ void tag1_fused_kernel(const float* __restrict__ x, const float* __restrict__ h1,
                                  const float* __restrict__ h2, const float* __restrict__ W,
                                  const float* __restrict__ b, float* __restrict__ gx,
                                  long long NV) {
    long long idx = (long long)blockIdx.x * blockDim.x + threadIdx.x;
    long long n = idx >> 6;
    int f = (int)(idx & 63);
    if (n >= NV) return;
    float acc = b[f];
    #pragma unroll
    for (int k = 0; k < 5; ++k) {
        acc += x[n * 5 + k] * W[k * 64 + f];
        acc += h1[n * 5 + k] * W[320 + k * 64 + f];
        acc += h2[n * 5 + k] * W[640 + k * 64 + f];
    }
    gx[n * 64 + f] = fmaxf(acc, 0.0f);
}

// ssum[n,f] /= max(cnt[n],1)
__global__ void mean_kernel(float* __restrict__ ssum, const float* __restrict__ cnt,
                            long long NS) {
    long long idx = (long long)blockIdx.x * blockDim.x + threadIdx.x;
    long long n = idx >> 6;
    if (n >= NS) return;
    ssum[idx] /= fmaxf(cnt[n], 1.0f);
}

// C[n,f] = relu(C[n,f] + b[f] + sum_k state_x[n,k]*W_root[k,f]),  k<6
__global__ void gc_epilogue_kernel(float* __restrict__ C, const float* __restrict__ sx,
                                   const float* __restrict__ Wroot, const float* __restrict__ b,
                                   long long NS) {
    long long idx = (long long)blockIdx.x * blockDim.x + threadIdx.x;
    long long n = idx >> 6;
    int f = (int)(idx & 63);
    if (n >= NS) return;
    float acc = C[idx] + b[f];
    #pragma unroll
    for (int k = 0; k < 6; ++k) acc += sx[n * 6 + k] * Wroot[k * 64 + f];
    C[idx] = fmaxf(acc, 0.0f);
}

__global__ void bias_relu_kernel(float* __restrict__ C, const float* __restrict__ b,
                                 long long NS) {
    long long idx = (long long)blockIdx.x * blockDim.x + threadIdx.x;
    if ((idx >> 6) >= NS) return;
    C[idx] = fmaxf(C[idx] + b[idx & 63], 0.0f);
}

// out[n,o] = sum_k sx[n,k]*W[k,o] + b[o],  W: [64,8]
__global__ void final_lin_kernel(const float* __restrict__ sx, const float* __restrict__ W,
                                 const float* __restrict__ b, float* __restrict__ out,
                                 long long NS) {
    long long idx = (long long)blockIdx.x * blockDim.x + threadIdx.x;
    long long n = idx >> 3;
    int o = (int)(idx & 7);
    if (n >= NS) return;
    float acc = b[o];
    #pragma unroll
    for (int k = 0; k < 64; ++k) acc += sx[n * 64 + k] * W[k * 8 + o];
    out[n * 8 + o] = acc;
}

// ---------------------------------------------------------------- host side

static inline unsigned grid1(long long total, int block) {
    return (unsigned)((total + block - 1) / block);
}

extern "C" void kernel_launch(void* const* d_in, const int* in_sizes, int n_in,
                              void* d_out, int out_size, void* d_ws, size_t ws_size,
                              hipStream_t stream) {
    const float* game_x   = (const float*)d_in[0];
    const float* state_x  = (const float*)d_in[1];
    const int*   ei_vv    = (const int*)d_in[2];
    const int*   ei_h     = (const int*)d_in[3];
    const float* eattr_h  = (const float*)d_in[4];
    const int*   ei_in    = (const int*)d_in[5];
    const int*   ei_ss    = (const int*)d_in[6];
    const float* tag1_W   = (const float*)d_in[7];
    const float* tag1_b   = (const float*)d_in[8];
    const float* tag2_W   = (const float*)d_in[9];
    const float* tag2_b   = (const float*)d_in[10];
    const float* gc_W_rel = (const float*)d_in[11];
    const float* gc_b     = (const float*)d_in[12];
    const float* gc_W_root= (const float*)d_in[13];
    const float* sage_W_l = (const float*)d_in[14];
    const float* sage_b_l = (const float*)d_in[15];
    const float* sage_W_r = (const float*)d_in[16];
    const float* lin_W    = (const float*)d_in[17];
    const float* lin_b    = (const float*)d_in[18];
    float* out = (float*)d_out;

    const long long NV  = in_sizes[0] / 5;
    const long long NS  = in_sizes[1] / 6;
    const long long EVV = in_sizes[2] / 2;
    const long long EH  = in_sizes[4];
    const long long EIN = in_sizes[5] / 2;
    const long long ESS = in_sizes[6] / 2;

    const int* src_vv = ei_vv;           const int* dst_vv = ei_vv + EVV;
    const int* src_h  = ei_h;            const int* dst_h  = ei_h  + EH;
    const int* src_in = ei_in;           const int* dst_in = ei_in + EIN;
    const int* src_ss = ei_ss;           const int* dst_ss = ei_ss + ESS;

    // workspace carve-up (all offsets 16B aligned: every count is a mult. of 4)
    float* p = (float*)d_ws;
    size_t off = 0;
    auto alloc = [&](size_t nf) { float* r = p + off; off += nf; return r; };
    float* gx     = alloc((size_t)NV * 64);  // later reused as tag2 output
    float* dinv_v = alloc((size_t)NV);
    float* h1v    = alloc((size_t)NV * 5);
    float* h2v    = alloc((size_t)NV * 5);
    float* agg    = alloc((size_t)NS * 64);  // gc aggregate; later tag2 hop1
    float* ssum   = alloc((size_t)NS * 64);  // sage sum/mean; later tag2 hop2
    float* cnt    = alloc((size_t)NS);
    float* dinv_s = alloc((size_t)NS);
    float* sxA    = alloc((size_t)NS * 64);  // gc out; later tag2 hop3
    float* sxB    = alloc((size_t)NS * 64);  // sage out
    float* tout   = gx;                      // tag2 out (NS*64 <= NV*64)
    (void)ws_size; (void)n_in; (void)out_size;

    const int B = 256;
    const unsigned gemmGrid = grid1(NS, 128);  // 8 waves x 16 rows per block

    // ---------------- TAGConv1 on (game_x, v->v), K=2 ----------------
    fill0_kernel<<<grid1(NV, B), B, 0, stream>>>(dinv_v, NV);
    fill0_kernel<<<grid1(NV * 5, B), B, 0, stream>>>(h1v, NV * 5);
    fill0_kernel<<<grid1(NV * 5, B), B, 0, stream>>>(h2v, NV * 5);
    degree_kernel<<<grid1(EVV, B), B, 0, stream>>>(dst_vv, dinv_v, EVV);
    rsqrt_deg_kernel<<<grid1(NV, B), B, 0, stream>>>(dinv_v, NV);
    scatter_kernel<5, 0><<<grid1(EVV * 5, B), B, 0, stream>>>(src_vv, dst_vv, game_x, h1v, dinv_v, EVV);
    scatter_kernel<5, 0><<<grid1(EVV * 5, B), B, 0, stream>>>(src_vv, dst_vv, h1v, h2v, dinv_v, EVV);
    tag1_fused_kernel<<<grid1(NV * 64, B), B, 0, stream>>>(game_x, h1v, h2v, tag1_W, tag1_b, gx, NV);

    // ---------------- GraphConv (history v->s, edge weights) ----------------
    fill0_kernel<<<grid1(NS * 64, B), B, 0, stream>>>(agg, NS * 64);
    scatter_kernel<64, 1><<<grid1(EH * 64, B), B, 0, stream>>>(src_h, dst_h, gx, agg, eattr_h, EH);
    gemm64_wmma<<<gemmGrid, B, 0, stream>>>(agg, gc_W_rel, sxA, (int)NS, 0);
    gc_epilogue_kernel<<<grid1(NS * 64, B), B, 0, stream>>>(sxA, state_x, gc_W_root, gc_b, NS);

    // ---------------- SAGEConv (in v->s, mean aggr) ----------------
    fill0_kernel<<<grid1(NS * 64, B), B, 0, stream>>>(ssum, NS * 64);
    fill0_kernel<<<grid1(NS, B), B, 0, stream>>>(cnt, NS);
    scatter_kernel<64, 2><<<grid1(EIN * 64, B), B, 0, stream>>>(src_in, dst_in, gx, ssum, (const float*)nullptr, EIN);
    degree_kernel<<<grid1(EIN, B), B, 0, stream>>>(dst_in, cnt, EIN);
    mean_kernel<<<grid1(NS * 64, B), B, 0, stream>>>(ssum, cnt, NS);
    gemm64_wmma<<<gemmGrid, B, 0, stream>>>(ssum, sage_W_l, sxB, (int)NS, 0);
    gemm64_wmma<<<gemmGrid, B, 0, stream>>>(sxA, sage_W_r, sxB, (int)NS, 1);
    bias_relu_kernel<<<grid1(NS * 64, B), B, 0, stream>>>(sxB, sage_b_l, NS);

    // ---------------- TAGConv2 on (sxB, s->s), K=3 ----------------
    fill0_kernel<<<grid1(NS, B), B, 0, stream>>>(dinv_s, NS);
    degree_kernel<<<grid1(ESS, B), B, 0, stream>>>(dst_ss, dinv_s, ESS);
    rsqrt_deg_kernel<<<grid1(NS, B), B, 0, stream>>>(dinv_s, NS);
    fill0_kernel<<<grid1(NS * 64, B), B, 0, stream>>>(agg, NS * 64);   // hop1
    scatter_kernel<64, 0><<<grid1(ESS * 64, B), B, 0, stream>>>(src_ss, dst_ss, sxB, agg, dinv_s, ESS);
    fill0_kernel<<<grid1(NS * 64, B), B, 0, stream>>>(ssum, NS * 64);  // hop2
    scatter_kernel<64, 0><<<grid1(ESS * 64, B), B, 0, stream>>>(src_ss, dst_ss, agg, ssum, dinv_s, ESS);
    fill0_kernel<<<grid1(NS * 64, B), B, 0, stream>>>(sxA, NS * 64);   // hop3
    scatter_kernel<64, 0><<<grid1(ESS * 64, B), B, 0, stream>>>(src_ss, dst_ss, ssum, sxA, dinv_s, ESS);
    gemm64_wmma<<<gemmGrid, B, 0, stream>>>(sxB,  tag2_W,          tout, (int)NS, 0);
    gemm64_wmma<<<gemmGrid, B, 0, stream>>>(agg,  tag2_W + 4096,   tout, (int)NS, 1);
    gemm64_wmma<<<gemmGrid, B, 0, stream>>>(ssum, tag2_W + 8192,   tout, (int)NS, 1);
    gemm64_wmma<<<gemmGrid, B, 0, stream>>>(sxA,  tag2_W + 12288,  tout, (int)NS, 1);
    bias_relu_kernel<<<grid1(NS * 64, B), B, 0, stream>>>(tout, tag2_b, NS);

    // ---------------- final linear [NS,64]@[64,8] ----------------
    final_lin_kernel<<<grid1(NS * 8, B), B, 0, stream>>>(tout, lin_W, lin_b, out, NS);
}